// fclip_23381801959769
// MI455X (gfx1250) — compile-verified
//
#include <hip/hip_runtime.h>
#include <hip/hip_bf16.h>
#include <math.h>

typedef __attribute__((ext_vector_type(2))) float v2f;
typedef __attribute__((ext_vector_type(8))) float v8f;

#define NB   128      // batch
#define TT   270      // time length
#define C0   270      // input channels
#define CB   320      // block channels
#define HARM 32
#define NCLS 1854
#define FEATC 128

__device__ __forceinline__ v8f wmma_f32(v2f a, v2f b, v8f c)
{
    return __builtin_amdgcn_wmma_f32_16x16x4_f32(false, a, false, b,
                                                 (short)0, c, false, false);
}

// ---------------------------------------------------------------------------
// 1) Spatial attention scale: scale[b] = sum_j softmax_j(maps[b,:]) (~1.0)
// ---------------------------------------------------------------------------
__global__ void attn_scale_kernel(const float* __restrict__ sloc,   // [128,2]
                                  const float* __restrict__ Wf,     // [270,32,32,2]
                                  float* __restrict__ scale)        // [128]
{
    const int b   = blockIdx.x;
    const int tid = threadIdx.x;          // blockDim.x == 320
    __shared__ float mm[4];
    __shared__ float ck[HARM], sk[HARM], cl[HARM], sl2[HARM];
    __shared__ float red[320];

    if (tid == 0) {
        float xmn = 1e30f, xmx = -1e30f, ymn = 1e30f, ymx = -1e30f;
        for (int n = 0; n < NB; ++n) {
            float x = sloc[2 * n + 0], y = sloc[2 * n + 1];
            xmn = fminf(xmn, x); xmx = fmaxf(xmx, x);
            ymn = fminf(ymn, y); ymx = fmaxf(ymx, y);
        }
        mm[0] = xmn; mm[1] = xmx; mm[2] = ymn; mm[3] = ymx;
    }
    __syncthreads();

    const float TWO_PI = 6.283185307179586f;
    float xb = (sloc[2 * b + 0] - mm[0]) / (mm[1] - mm[0]);
    float yb = (sloc[2 * b + 1] - mm[2]) / (mm[3] - mm[2]);
    if (tid < HARM) {
        float a = TWO_PI * (float)tid * xb;
        ck[tid] = cosf(a); sk[tid] = sinf(a);
    } else if (tid < 2 * HARM) {
        int l = tid - HARM;
        float a = TWO_PI * (float)l * yb;
        cl[l] = cosf(a); sl2[l] = sinf(a);
    }
    __syncthreads();

    const int j = tid;
    float m = 0.0f;
    if (j < C0) {
        const float* wj = Wf + (size_t)j * HARM * HARM * 2;
        for (int k = 0; k < HARM; ++k) {
            float accC = 0.0f, accS = 0.0f;
            for (int l = 0; l < HARM; ++l) {
                float w0 = wj[(k * HARM + l) * 2 + 0];
                float w1 = wj[(k * HARM + l) * 2 + 1];
                accC += cl[l] * w0 + sl2[l] * w1;
                accS += cl[l] * w1 - sl2[l] * w0;
            }
            m += ck[k] * accC + sk[k] * accS;
        }
    }
    red[tid] = (j < C0) ? m : -1e30f;
    __syncthreads();
    for (int s = 256; s > 0; s >>= 1) {
        if (tid < s && tid + s < 320) red[tid] = fmaxf(red[tid], red[tid + s]);
        __syncthreads();
    }
    float mx = red[0];
    __syncthreads();
    float e = (j < C0) ? expf(m - mx) : 0.0f;
    red[tid] = e;
    __syncthreads();
    for (int s = 256; s > 0; s >>= 1) {
        if (tid < s && tid + s < 320) red[tid] += red[tid + s];
        __syncthreads();
    }
    float total = red[0];
    __syncthreads();
    red[tid] = (j < C0) ? (e / total) : 0.0f;
    __syncthreads();
    for (int s = 256; s > 0; s >>= 1) {
        if (tid < s && tid + s < 320) red[tid] += red[tid + s];
        __syncthreads();
    }
    if (tid == 0) scale[b] = red[0];
}

// ---------------------------------------------------------------------------
// 2) Subject mixing: Y[b,c,t] = scale[b] * sum_j X[b,c,j] * M[sid[b],j,t]
//    Both operands staged in LDS (zero-filled) -> branch-free WMMA loop.
//    Block: 128 thr, tile M=32 x N=64, K chunks of 32.
// ---------------------------------------------------------------------------
__global__ __launch_bounds__(128)
void mix_wmma_kernel(const float* __restrict__ X,
                     const float* __restrict__ SM,
                     const int*   __restrict__ sid,
                     const float* __restrict__ scale,
                     float* __restrict__ Y)
{
    constexpr int ASTR = 36;   // LDS stride for A tile [32 rows(M) x 32 cols(K)]
    constexpr int BSTR = 72;   // LDS stride for B tile [32 rows(K) x 64 cols(N)]
    __shared__ float As[32 * ASTR];
    __shared__ float Bs[32 * BSTR];

    const int b    = blockIdx.z;
    const int m0   = blockIdx.y * 32;
    const int n0b  = blockIdx.x * 64;
    const int tid  = threadIdx.x;
    const int wave = tid >> 5;
    const int lane = tid & 31;
    const int l16  = lane & 15;
    const int half = lane >> 4;
    const int colb = wave * 16 + l16;

    const float* Xb = X  + (size_t)b * C0 * TT;
    const float* Mb = SM + (size_t)sid[b] * C0 * TT;

    const int srow = tid >> 2;           // 32 rows, 4 threads/row
    const int ac0  = (tid & 3) * 8;      // 8 A-cols per thread
    const int bc0  = (tid & 3) * 16;     // 16 B-cols per thread

    v8f acc0 = {0,0,0,0,0,0,0,0};
    v8f acc1 = {0,0,0,0,0,0,0,0};

    for (int ch = 0; ch < 9; ++ch) {     // ceil(270/32)
        const int k0 = ch * 32;
        // stage A: As[r][c] = X[b, m0+r, k0+c] (zero-filled)
        {
            const int r = m0 + srow;
            #pragma unroll
            for (int jj = 0; jj < 8; ++jj) {
                int c = k0 + ac0 + jj;
                As[srow * ASTR + ac0 + jj] =
                    (r < C0 && c < C0) ? Xb[(size_t)r * TT + c] : 0.0f;
            }
        }
        // stage B: Bs[r][c] = M[k0+r, n0b+c] (zero-filled)
        {
            const int r = k0 + srow;
            #pragma unroll
            for (int jj = 0; jj < 16; ++jj) {
                int c = n0b + bc0 + jj;
                Bs[srow * BSTR + bc0 + jj] =
                    (r < C0 && c < TT) ? Mb[(size_t)r * TT + c] : 0.0f;
            }
        }
        __syncthreads();

        #pragma unroll
        for (int s = 0; s < 8; ++s) {
            const int lka = s * 4 + half * 2;
            v2f a0, a1, bb;
            a0.x = As[l16 * ASTR + lka];
            a0.y = As[l16 * ASTR + lka + 1];
            a1.x = As[(16 + l16) * ASTR + lka];
            a1.y = As[(16 + l16) * ASTR + lka + 1];
            bb.x = Bs[lka * BSTR + colb];
            bb.y = Bs[(lka + 1) * BSTR + colb];
            acc0 = wmma_f32(a0, bb, acc0);
            acc1 = wmma_f32(a1, bb, acc1);
        }
        __syncthreads();
    }

    const float sc = scale[b];
    const int bcol = n0b + colb;
    if (bcol < TT) {
        #pragma unroll
        for (int r = 0; r < 8; ++r) {
            int row0 = m0 + half * 8 + r;
            int row1 = row0 + 16;
            if (row0 < C0) Y[((size_t)b * C0 + row0) * TT + bcol] = acc0[r] * sc;
            if (row1 < C0) Y[((size_t)b * C0 + row1) * TT + bcol] = acc1[r] * sc;
        }
    }
}

// ---------------------------------------------------------------------------
// 3) Implicit-GEMM dilated conv1d via WMMA f32 16x16x4.
//    A = W[Cout, Cin*TAPS] (row-major weights), B staged in LDS with dilation
//    halo (+-4) and zero-fill. Block: M=32 x N=64, 32 input channels/stage.
// ---------------------------------------------------------------------------
template <int TAPS>
__device__ __forceinline__ float lds_b(const float* Bs, int lka, int colbase, int dil)
{
    if (TAPS == 1) {
        return Bs[lka * 72 + colbase];
    } else {
        int i   = lka / 3;
        int tap = lka - 3 * i;
        return Bs[i * 72 + colbase + (tap - 1) * dil];
    }
}

template <int TAPS>
__global__ __launch_bounds__(128)
void conv_wmma_kernel(const float* __restrict__ X,
                      const float* __restrict__ W,
                      const float* __restrict__ bias,
                      float* __restrict__ Y,
                      int Cin, int Cout, int dil)
{
    constexpr int CHUNK = 32;            // input channels per stage
    constexpr int BSTR  = 72;            // 64 cols + 4 halo each side
    __shared__ float Bs[CHUNK * BSTR];   // 9216 B

    const int b    = blockIdx.z;
    const int m0   = blockIdx.y * 32;
    const int n0b  = blockIdx.x * 64;
    const int tid  = threadIdx.x;
    const int wave = tid >> 5;
    const int lane = tid & 31;
    const int l16  = lane & 15;
    const int half = lane >> 4;
    const int K    = Cin * TAPS;
    const int colbase = wave * 16 + l16 + 4;

    const float* Xb    = X + (size_t)b * Cin * TT;
    const float* Arow0 = W + (size_t)(m0 + l16) * K;
    const float* Arow1 = W + (size_t)(m0 + 16 + l16) * K;

    const int srowL = tid >> 2;          // 32 rows, 4 threads/row
    const int sc0   = (tid & 3) * 18;    // 18 cols per thread (72 total)
    const bool interior = (n0b >= 4) && (n0b + 68 <= TT);

    v8f acc0 = {0,0,0,0,0,0,0,0};
    v8f acc1 = {0,0,0,0,0,0,0,0};

    const int nchunk = (Cin + CHUNK - 1) / CHUNK;
    for (int ch = 0; ch < nchunk; ++ch) {
        // ---- stage B chunk (zero-filled halo/boundaries) ----
        {
            const int srow = ch * CHUNK + srowL;
            const float* xr = Xb + (size_t)srow * TT;
            const bool fullrows = ((ch + 1) * CHUNK <= Cin);
            if (fullrows && interior) {
                #pragma unroll
                for (int jj = 0; jj < 18; ++jj)
                    Bs[srowL * BSTR + sc0 + jj] = xr[n0b - 4 + sc0 + jj];
            } else {
                const bool rok = (srow < Cin);
                #pragma unroll
                for (int jj = 0; jj < 18; ++jj) {
                    int t = n0b - 4 + sc0 + jj;
                    Bs[srowL * BSTR + sc0 + jj] =
                        (rok && t >= 0 && t < TT) ? xr[t] : 0.0f;
                }
            }
        }
        __syncthreads();

        const int kkbase = ch * (CHUNK * TAPS);
        const int kkrem  = K - kkbase;
        if (kkrem >= CHUNK * TAPS) {
            // -------- branch-free steady state --------
            #pragma unroll
            for (int s = 0; s < (CHUNK * TAPS) / 4; ++s) {
                const int lka = s * 4 + half * 2;
                v2f a0, a1, bb;
                a0.x = Arow0[kkbase + lka];
                a0.y = Arow0[kkbase + lka + 1];
                a1.x = Arow1[kkbase + lka];
                a1.y = Arow1[kkbase + lka + 1];
                bb.x = lds_b<TAPS>(Bs, lka,     colbase, dil);
                bb.y = lds_b<TAPS>(Bs, lka + 1, colbase, dil);
                acc0 = wmma_f32(a0, bb, acc0);
                acc1 = wmma_f32(a1, bb, acc1);
            }
        } else {
            // -------- tail chunk (Cin=270 only): guard A, B is zero-filled ----
            const int steps = (kkrem + 3) / 4;
            for (int s = 0; s < steps; ++s) {
                const int lka = s * 4 + half * 2;
                const int ka  = kkbase + lka;
                v2f a0, a1, bb;
                a0.x = (ka     < K) ? Arow0[ka]     : 0.0f;
                a0.y = (ka + 1 < K) ? Arow0[ka + 1] : 0.0f;
                a1.x = (ka     < K) ? Arow1[ka]     : 0.0f;
                a1.y = (ka + 1 < K) ? Arow1[ka + 1] : 0.0f;
                bb.x = lds_b<TAPS>(Bs, lka,     colbase, dil);
                bb.y = lds_b<TAPS>(Bs, lka + 1, colbase, dil);
                acc0 = wmma_f32(a0, bb, acc0);
                acc1 = wmma_f32(a1, bb, acc1);
            }
        }
        __syncthreads();
    }

    const int bcol = n0b + wave * 16 + l16;
    if (bcol < TT) {
        #pragma unroll
        for (int r = 0; r < 8; ++r) {
            int row0 = m0 + half * 8 + r;
            int row1 = row0 + 16;
            Y[((size_t)b * Cout + row0) * TT + bcol] = acc0[r] + bias[row0];
            Y[((size_t)b * Cout + row1) * TT + bcol] = acc1[r] + bias[row1];
        }
    }
}

// ---------------------------------------------------------------------------
// 4) BatchNorm statistics (training mode, biased var over batch+time).
// ---------------------------------------------------------------------------
__global__ void bn_stats_kernel(const float* __restrict__ Y, int C,
                                float* __restrict__ mean,
                                float* __restrict__ invstd)
{
    const int c   = blockIdx.x;
    const int tid = threadIdx.x;                  // 256
    __shared__ float s1[256], s2[256];
    float a = 0.0f, q = 0.0f;
    const int N = NB * TT;
    for (int i = tid; i < N; i += 256) {
        int b = i / TT, t = i - b * TT;
        float v = Y[((size_t)b * C + c) * TT + t];
        a += v; q += v * v;
    }
    s1[tid] = a; s2[tid] = q;
    __syncthreads();
    for (int s = 128; s > 0; s >>= 1) {
        if (tid < s) { s1[tid] += s1[tid + s]; s2[tid] += s2[tid + s]; }
        __syncthreads();
    }
    if (tid == 0) {
        float m = s1[0] / (float)N;
        float v = s2[0] / (float)N - m * m;
        mean[c]   = m;
        invstd[c] = rsqrtf(v + 1e-5f);
    }
}

// ---------------------------------------------------------------------------
// 5) BN apply (+ exact GELU), elementwise in-place.
// ---------------------------------------------------------------------------
__device__ __forceinline__ float gelu_exact(float x)
{
    return 0.5f * x * (1.0f + erff(x * 0.70710678118654752f));
}

template <bool GELU>
__global__ void bn_act_kernel(float* __restrict__ Y, int C,
                              const float* __restrict__ mean,
                              const float* __restrict__ invstd,
                              const float* __restrict__ g,
                              const float* __restrict__ be)
{
    const size_t i = (size_t)blockIdx.x * blockDim.x + threadIdx.x;
    const size_t total = (size_t)NB * C * TT;
    if (i >= total) return;
    int c = (int)((i / TT) % C);
    float v = (Y[i] - mean[c]) * invstd[c] * g[c] + be[c];
    Y[i] = GELU ? gelu_exact(v) : v;
}

// ---------------------------------------------------------------------------
// 6) BN + GLU: in [B,640,T] -> out [B,320,T]; g/be duplicated across halves.
// ---------------------------------------------------------------------------
__global__ void bn_glu_kernel(const float* __restrict__ Yin,
                              float* __restrict__ Yout,
                              const float* __restrict__ mean,
                              const float* __restrict__ invstd,
                              const float* __restrict__ g,
                              const float* __restrict__ be)
{
    const size_t i = (size_t)blockIdx.x * blockDim.x + threadIdx.x;
    const size_t total = (size_t)NB * CB * TT;
    if (i >= total) return;
    int t = (int)(i % TT);
    int c = (int)((i / TT) % CB);
    int b = (int)(i / ((size_t)TT * CB));
    float u = Yin[((size_t)b * (2 * CB) + c) * TT + t];
    float w = Yin[((size_t)b * (2 * CB) + c + CB) * TT + t];
    u = (u - mean[c])      * invstd[c]      * g[c] + be[c];
    w = (w - mean[c + CB]) * invstd[c + CB] * g[c] + be[c];
    Yout[((size_t)b * CB + c) * TT + t] = u * (1.0f / (1.0f + expf(-w)));
}

// ---------------------------------------------------------------------------
// 7) Plain GELU elementwise (final wf1 path, no BN).
// ---------------------------------------------------------------------------
__global__ void gelu_kernel(float* __restrict__ Y, size_t total)
{
    const size_t i = (size_t)blockIdx.x * blockDim.x + threadIdx.x;
    if (i < total) Y[i] = gelu_exact(Y[i]);
}

// ---------------------------------------------------------------------------
// 8) Mean over time: [B,128,T] -> [B,128]
// ---------------------------------------------------------------------------
__global__ void mean_time_kernel(const float* __restrict__ Y,
                                 float* __restrict__ feat)
{
    const int i = blockIdx.x * blockDim.x + threadIdx.x;
    if (i >= NB * FEATC) return;
    int b = i / FEATC, c = i - b * FEATC;
    const float* p = Y + ((size_t)b * FEATC + c) * TT;
    float s = 0.0f;
    for (int t = 0; t < TT; ++t) s += p[t];
    feat[i] = s * (1.0f / (float)TT);
}

// ---------------------------------------------------------------------------
// 9) Classifier: out[b,n] = feat[b,:] . cw[n,:] + cb[n]
// ---------------------------------------------------------------------------
__global__ void classifier_kernel(const float* __restrict__ feat,
                                  const float* __restrict__ cw,
                                  const float* __restrict__ cb,
                                  float* __restrict__ out)
{
    const int i = blockIdx.x * blockDim.x + threadIdx.x;
    if (i >= NB * NCLS) return;
    int b = i / NCLS, n = i - b * NCLS;
    const float* f = feat + (size_t)b * FEATC;
    const float* w = cw + (size_t)n * FEATC;
    float s = cb[n];
    for (int c = 0; c < FEATC; ++c) s += f[c] * w[c];
    out[i] = s;
}

// ---------------------------------------------------------------------------
// Host-side orchestration
// ---------------------------------------------------------------------------
extern "C" void kernel_launch(void* const* d_in, const int* in_sizes, int n_in,
                              void* d_out, int out_size, void* d_ws, size_t ws_size,
                              hipStream_t stream)
{
    (void)in_sizes; (void)n_in; (void)out_size; (void)ws_size;

    const float* X     = (const float*)d_in[0];   // [128,270,270]
    const int*   sid   = (const int*)  d_in[1];   // [128]
    const float* sloc  = (const float*)d_in[2];   // [128,2]
    const float* Wf    = (const float*)d_in[3];   // [270,32,32,2]
    const float* SM    = (const float*)d_in[4];   // [4,270,270]
    const float* wf1 = (const float*)d_in[45];    // [640,320,1]
    const float* bf1 = (const float*)d_in[46];
    const float* wf2 = (const float*)d_in[47];    // [128,640,1]
    const float* bf2 = (const float*)d_in[48];
    const float* cw  = (const float*)d_in[49];    // [1854,128]
    const float* cb  = (const float*)d_in[50];
    float* out = (float*)d_out;                   // [128,1854]

    const size_t bufElems = (size_t)NB * 640 * TT;
    float* bufA   = (float*)d_ws;
    float* bufB   = bufA + bufElems;
    float* mean   = bufB + bufElems;
    float* invstd = mean + 640;
    float* scale  = invstd + 640;
    float* feat   = scale + NB;

    // 1) attention scale
    attn_scale_kernel<<<NB, 320, 0, stream>>>(sloc, Wf, scale);

    // 2) subject mixing -> bufA as [128,270,270]
    {
        dim3 grid((TT + 63) / 64, (C0 + 31) / 32, NB);
        mix_wmma_kernel<<<grid, 128, 0, stream>>>(X, SM, sid, scale, bufA);
    }

    const int ntile = (TT + 63) / 64;
    auto conv3 = [&](const float* src, const float* w, const float* bi,
                     float* dst, int cin, int cout, int dil) {
        dim3 grid(ntile, cout / 32, NB);
        conv_wmma_kernel<3><<<grid, 128, 0, stream>>>(src, w, bi, dst, cin, cout, dil);
    };
    auto conv1x1 = [&](const float* src, const float* w, const float* bi,
                       float* dst, int cin, int cout) {
        dim3 grid(ntile, cout / 32, NB);
        conv_wmma_kernel<1><<<grid, 128, 0, stream>>>(src, w, bi, dst, cin, cout, 1);
    };
    auto stats = [&](const float* y, int c) {
        bn_stats_kernel<<<c, 256, 0, stream>>>(y, c, mean, invstd);
    };
    auto bngelu = [&](float* y, int c, const float* g, const float* be) {
        size_t total = (size_t)NB * c * TT;
        bn_act_kernel<true><<<(int)((total + 255) / 256), 256, 0, stream>>>(
            y, c, mean, invstd, g, be);
    };

    // 3) residual blocks
    for (int i = 0; i < 5; ++i) {
        const int base = 5 + i * 8;
        const float* w1 = (const float*)d_in[base + 0];
        const float* b1 = (const float*)d_in[base + 1];
        const float* w2 = (const float*)d_in[base + 2];
        const float* b2 = (const float*)d_in[base + 3];
        const float* w3 = (const float*)d_in[base + 4];
        const float* b3 = (const float*)d_in[base + 5];
        const float* g  = (const float*)d_in[base + 6];
        const float* be = (const float*)d_in[base + 7];
        const int cin = (i == 0) ? C0 : CB;

        conv3(bufA, w1, b1, bufB, cin, CB, 1);      // conv1 (dil 1)
        stats(bufB, CB);
        bngelu(bufB, CB, g, be);

        conv3(bufB, w2, b2, bufA, CB, CB, 2);       // conv2 (dil 2)
        stats(bufA, CB);
        bngelu(bufA, CB, g, be);

        conv3(bufA, w3, b3, bufB, CB, 2 * CB, 4);   // conv3 (dil 4) -> 640ch
        stats(bufB, 2 * CB);
        {
            size_t total = (size_t)NB * CB * TT;
            bn_glu_kernel<<<(int)((total + 255) / 256), 256, 0, stream>>>(
                bufB, bufA, mean, invstd, g, be);   // -> bufA [128,320,270]
        }
    }

    // 4) head: wf1 (1x1, 320->640) + GELU
    conv1x1(bufA, wf1, bf1, bufB, CB, 640);
    {
        size_t total = (size_t)NB * 640 * TT;
        gelu_kernel<<<(int)((total + 255) / 256), 256, 0, stream>>>(bufB, total);
    }
    // wf2 (1x1, 640->128)
    conv1x1(bufB, wf2, bf2, bufA, 640, FEATC);
    // mean over time
    mean_time_kernel<<<(NB * FEATC + 255) / 256, 256, 0, stream>>>(bufA, feat);
    // classifier
    classifier_kernel<<<(NB * NCLS + 255) / 256, 256, 0, stream>>>(feat, cw, cb, out);
}